// NewtonNet_32203664785943
// MI455X (gfx1250) — compile-verified
//
#include <hip/hip_runtime.h>
#include <hip/hip_bf16.h>

#define BATCH 4
#define NATOM 512
#define NNBR  32
#define FDIM  128
#define NBAS  20
#define NLAY  3
#define LSTR  132   // padded LDS row stride (floats) -> conflict-free A reads

typedef __attribute__((ext_vector_type(2))) float v2f;
typedef __attribute__((ext_vector_type(8))) float v8f;

// ---------------------------------------------------------------------------
// 32x128 = (32x128 A in LDS) @ (128x128 W row-major in global), fp32 WMMA.
// 8 waves: wave w owns col-tile w and BOTH row tiles (B fragment shared by
// the two row-tile WMMAs; weight matrix is read exactly once per block).
// Caller must __syncthreads() before (A ready) and after (Out ready).
// ---------------------------------------------------------------------------
__device__ __forceinline__ void gemm32x128(const float* __restrict__ W,
                                           const float* __restrict__ Alds,
                                           float* __restrict__ Out, int ostride,
                                           const float* __restrict__ bias,
                                           bool do_silu, int tid)
{
    const int wave = tid >> 5;       // 0..7 -> col tile
    const int lane = tid & 31;
    const int rc   = lane & 15;      // row (A/D) or col (B/D) within tile
    const int kh   = lane >> 4;      // K-half selector

    v8f acc0 = {0.f,0.f,0.f,0.f,0.f,0.f,0.f,0.f};   // rows 0..15
    v8f acc1 = acc0;                                 // rows 16..31

    const float* arow0 = Alds + rc * LSTR + kh * 2;           // row tile 0
    const float* arow1 = Alds + (16 + rc) * LSTR + kh * 2;    // row tile 1
    const float* wcol  = W + wave * 16 + rc;                  // this wave's col

    #pragma unroll 4
    for (int k0 = 0; k0 < FDIM; k0 += 4) {
        const int kr = (k0 + kh * 2) << 7;   // *128
        v2f b;  b.x  = wcol[kr];    b.y  = wcol[kr + FDIM];
        v2f a0; a0.x = arow0[k0];   a0.y = arow0[k0 + 1];
        v2f a1; a1.x = arow1[k0];   a1.y = arow1[k0 + 1];
        acc0 = __builtin_amdgcn_wmma_f32_16x16x4_f32(false, a0, false, b,
                                                     (short)0, acc0, false, false);
        acc1 = __builtin_amdgcn_wmma_f32_16x16x4_f32(false, a1, false, b,
                                                     (short)0, acc1, false, false);
    }

    const int c = wave * 16 + rc;
    const float bb = bias ? bias[c] : 0.f;
    #pragma unroll
    for (int i = 0; i < 8; ++i) {
        const int r = kh * 8 + i;
        float v0 = acc0[i] + bb;      // row r      (tile 0)
        float v1 = acc1[i] + bb;      // row 16 + r (tile 1)
        if (do_silu) {
            v0 = v0 / (1.f + __expf(-v0));
            v1 = v1 / (1.f + __expf(-v1));
        }
        Out[r * ostride + c]        = v0;
        Out[(16 + r) * ostride + c] = v1;
    }
}

// ---------------------------------------------------------------------------
// init: inv = emb[z]; zero eqF/eqf/eqdr; geometry: dvec, cutoff, cut*bessel
// ---------------------------------------------------------------------------
__global__ void __launch_bounds__(128)
k_init(const int* __restrict__ z, const float* __restrict__ pos,
       const int* __restrict__ nbrs, const float* __restrict__ emb,
       float* __restrict__ inv, float* __restrict__ eqF, float* __restrict__ eqf,
       float* __restrict__ eqdr0, float* __restrict__ eqdr1,
       float* __restrict__ edge, float* __restrict__ cut, float* __restrict__ dvec)
{
    const int g = blockIdx.x;            // atom index b*NATOM + a
    const int b = g >> 9;
    const int t = threadIdx.x;

    if (t < FDIM) {
        inv[g * FDIM + t] = emb[z[g] * FDIM + t];
        #pragma unroll
        for (int d = 0; d < 3; ++d) {
            eqf  [(g * 3 + d) * FDIM + t] = 0.f;
            eqdr0[(g * 3 + d) * FDIM + t] = 0.f;
            eqdr1[(g * 3 + d) * FDIM + t] = 0.f;
        }
    }
    if (t < 3) eqF[g * 3 + t] = 0.f;
    if (t < NNBR) {
        const int n  = t;
        const int gj = b * NATOM + nbrs[g * NNBR + n];
        const float dx = pos[g * 3 + 0] - pos[gj * 3 + 0];
        const float dy = pos[g * 3 + 1] - pos[gj * 3 + 1];
        const float dz = pos[g * 3 + 2] - pos[gj * 3 + 2];
        const float dist = sqrtf(dx * dx + dy * dy + dz * dz + 1e-12f);
        dvec[(g * NNBR + n) * 3 + 0] = dx;
        dvec[(g * NNBR + n) * 3 + 1] = dy;
        dvec[(g * NNBR + n) * 3 + 2] = dz;
        const float x = dist * 0.2f;     // dist / 5.0
        float c = 0.f;
        if (x < 1.f) {
            const float x2 = x * x, x3 = x2 * x, x4 = x2 * x2, x5 = x4 * x;
            c = 1.f - 6.f * x5 + 15.f * x4 - 10.f * x3;
        }
        cut[g * NNBR + n] = c;
        const float dsafe = dist > 1e-8f ? dist : 1.f;
        const float s = sqrtf(2.f / 5.f) / dsafe;
        for (int k = 1; k <= NBAS; ++k) {
            float v = s * sinf((float)k * 3.14159265358979323846f * dsafe / 5.f);
            if (dist <= 1e-8f) v = 0.f;
            edge[(g * NNBR + n) * NBAS + (k - 1)] = c * v;   // inv_edge = cut*bessel
        }
    }
}

// ---------------------------------------------------------------------------
// imn = silu(inv @ W1 + b1) @ W2 + b2   (all atoms; 32 rows per block, WMMA)
// ---------------------------------------------------------------------------
__global__ void __launch_bounds__(256)
k_imn(const float* __restrict__ inv, const float* __restrict__ W1,
      const float* __restrict__ b1, const float* __restrict__ W2,
      const float* __restrict__ b2, float* __restrict__ imn)
{
    __shared__ float sA[32 * LSTR];
    __shared__ float sH[32 * LSTR];
    const int t = threadIdx.x;
    const int base = blockIdx.x * 32;
    for (int i = t; i < 32 * FDIM; i += 256) {
        const int r = i >> 7, c = i & 127;
        sA[r * LSTR + c] = inv[(base + r) * FDIM + c];
    }
    __syncthreads();
    gemm32x128(W1, sA, sH, LSTR, b1, true, t);
    __syncthreads();
    gemm32x128(W2, sH, imn + (size_t)base * FDIM, FDIM, b2, false, t);
}

// ---------------------------------------------------------------------------
// fused per-atom layer body (one block = one atom, 256 threads = 8 waves)
// ---------------------------------------------------------------------------
__global__ void __launch_bounds__(256)
k_layer(const int* __restrict__ nbrs, const float* __restrict__ mask,
        const float* __restrict__ edge, const float* __restrict__ cutb,
        const float* __restrict__ dvecb, const float* __restrict__ imn,
        float* __restrict__ inv, float* __restrict__ eqF, float* __restrict__ eqf,
        const float* __restrict__ eqdr_in, float* __restrict__ eqdr_out,
        const float* __restrict__ W_ime, const float* __restrict__ b_ime,
        const float* __restrict__ W_emc,
        const float* __restrict__ W_emf1, const float* __restrict__ b_emf1,
        const float* __restrict__ W_emf2, const float* __restrict__ b_emf2,
        const float* __restrict__ W_eme1, const float* __restrict__ W_eme2,
        const float* __restrict__ W_esc1, const float* __restrict__ b_esc1,
        const float* __restrict__ W_esc2, const float* __restrict__ b_esc2,
        const float* __restrict__ W_isc1, const float* __restrict__ b_isc1,
        const float* __restrict__ W_isc2, const float* __restrict__ b_isc2,
        const float* __restrict__ ln_g, const float* __restrict__ ln_b)
{
    __shared__ float s_msg [32 * LSTR];
    __shared__ float s_buf [32 * LSTR];   // imn_nbr, then MLP hidden
    __shared__ float s_feat[32 * LSTR];   // feat, then eme
    __shared__ float s_imns[FDIM];
    __shared__ float s_edge[32 * NBAS];
    __shared__ float s_cut[32], s_mask[32], s_c[32];
    __shared__ float s_dvec[32 * 3];
    __shared__ int   s_nbr[32];
    __shared__ float s_updf [3 * FDIM];
    __shared__ float s_eqfn [3 * FDIM];
    __shared__ float s_eqdrn[3 * FDIM];
    __shared__ float s_inv[FDIM], s_esc[FDIM], s_isc[FDIM], s_h[FDIM], s_red[FDIM];
    __shared__ float s_acc[2 * 3 * FDIM];

    const int g = blockIdx.x;
    const int b = g >> 9;
    const int t = threadIdx.x;

    // ---- loads ----
    if (t < FDIM) s_imns[t] = imn[g * FDIM + t];
    if (t < NNBR) {
        s_nbr [t] = nbrs[g * NNBR + t];
        s_cut [t] = cutb[g * NNBR + t];
        s_mask[t] = mask[g * NNBR + t];
    }
    if (t < 96) s_dvec[t] = dvecb[g * 96 + t];
    for (int i = t; i < 32 * NBAS; i += 256) s_edge[i] = edge[g * (NNBR * NBAS) + i];
    __syncthreads();
    for (int i = t; i < 32 * FDIM; i += 256) {          // gather neighbor imn
        const int n = i >> 7, f = i & 127;
        s_buf[n * LSTR + f] = imn[((size_t)b * NATOM + s_nbr[n]) * FDIM + f];
    }
    __syncthreads();

    // ---- msg = (inv_edge@W_ime + b)*cut * imn_self * imn_nbr ----
    for (int i = t; i < 32 * FDIM; i += 256) {
        const int n = i >> 7, f = i & 127;
        float acc = b_ime[f];
        #pragma unroll
        for (int k = 0; k < NBAS; ++k) acc += s_edge[n * NBAS + k] * W_ime[k * FDIM + f];
        s_msg[n * LSTR + f] = acc * s_cut[n] * s_imns[f] * s_buf[n * LSTR + f];
    }
    __syncthreads();

    // ---- inv1 = inv + sum_n msg*mask ; c[n] = msg[n] . W_emc ----
    if (t < FDIM) {
        float acc = inv[g * FDIM + t];
        for (int n = 0; n < NNBR; ++n) acc += s_msg[n * LSTR + t] * s_mask[n];
        s_inv[t] = acc;
    } else if (t < 160) {
        const int n = t - 128;
        float acc = 0.f;
        for (int f = 0; f < FDIM; ++f) acc += s_msg[n * LSTR + f] * W_emc[f];
        s_c[n] = acc;
    }
    __syncthreads();

    if (t < 3) {    // eqF += sum_n mask * c * dvec
        float acc = 0.f;
        for (int n = 0; n < NNBR; ++n) acc += s_mask[n] * s_c[n] * s_dvec[n * 3 + t];
        eqF[g * 3 + t] += acc;
    }

    // ---- feat = silu(msg@W_emf1+b)@W_emf2+b ----
    gemm32x128(W_emf1, s_msg, s_buf, LSTR, b_emf1, true, t);
    __syncthreads();
    gemm32x128(W_emf2, s_buf, s_feat, LSTR, b_emf2, false, t);
    __syncthreads();

    // ---- upd_f[d][f] = sum_n mask*c[n]*feat[n][f]*dvec[n][d]; eqf += ----
    {
        const int f = t & 127, h = t >> 7;
        float a0 = 0.f, a1 = 0.f, a2 = 0.f;
        for (int n = h * 16; n < h * 16 + 16; ++n) {
            const float w = s_mask[n] * s_c[n] * s_feat[n * LSTR + f];
            a0 += w * s_dvec[n * 3 + 0];
            a1 += w * s_dvec[n * 3 + 1];
            a2 += w * s_dvec[n * 3 + 2];
        }
        s_acc[(h * 3 + 0) * FDIM + f] = a0;
        s_acc[(h * 3 + 1) * FDIM + f] = a1;
        s_acc[(h * 3 + 2) * FDIM + f] = a2;
    }
    __syncthreads();
    if (t < FDIM) {
        #pragma unroll
        for (int d = 0; d < 3; ++d) {
            const float u = s_acc[d * FDIM + t] + s_acc[(3 + d) * FDIM + t];
            s_updf[d * FDIM + t] = u;
            const float e = eqf[((size_t)g * 3 + d) * FDIM + t] + u;
            s_eqfn[d * FDIM + t] = e;
            eqf[((size_t)g * 3 + d) * FDIM + t] = e;
        }
    }
    __syncthreads();

    // ---- eme = silu(msg@W_eme1)@W_eme2 (no bias) ----
    gemm32x128(W_eme1, s_msg, s_buf, LSTR, nullptr, true, t);
    __syncthreads();
    gemm32x128(W_eme2, s_buf, s_feat, LSTR, nullptr, false, t);
    __syncthreads();

    // ---- eqdr delta: sum_n mask * eme[n][f] * eqdr_in[nbr[n]][d][f] ----
    {
        const int f = t & 127, h = t >> 7;
        float a0 = 0.f, a1 = 0.f, a2 = 0.f;
        for (int n = h * 16; n < h * 16 + 16; ++n) {
            const float w = s_mask[n] * s_feat[n * LSTR + f];
            const float* dr = eqdr_in + ((size_t)b * NATOM + s_nbr[n]) * (3 * FDIM);
            a0 += w * dr[f];
            a1 += w * dr[FDIM + f];
            a2 += w * dr[2 * FDIM + f];
        }
        s_acc[(h * 3 + 0) * FDIM + f] = a0;
        s_acc[(h * 3 + 1) * FDIM + f] = a1;
        s_acc[(h * 3 + 2) * FDIM + f] = a2;
    }
    __syncthreads();

    // ---- esc = silu(inv1@W_esc1+b)@W_esc2+b (VALU) ----
    if (t < FDIM) {
        float acc = b_esc1[t];
        for (int k = 0; k < FDIM; ++k) acc += s_inv[k] * W_esc1[k * FDIM + t];
        s_h[t] = acc / (1.f + __expf(-acc));
    }
    __syncthreads();
    if (t < FDIM) {
        float acc = b_esc2[t];
        for (int k = 0; k < FDIM; ++k) acc += s_h[k] * W_esc2[k * FDIM + t];
        s_esc[t] = acc;
    }
    __syncthreads();

    // ---- eqdr_new = eqdr_in[self] + delta + esc*upd_f ----
    if (t < FDIM) {
        #pragma unroll
        for (int d = 0; d < 3; ++d) {
            const float v = eqdr_in[((size_t)g * 3 + d) * FDIM + t]
                          + s_acc[d * FDIM + t] + s_acc[(3 + d) * FDIM + t]
                          + s_esc[t] * s_updf[d * FDIM + t];
            s_eqdrn[d * FDIM + t] = v;
            eqdr_out[((size_t)g * 3 + d) * FDIM + t] = v;
        }
    }

    // ---- isc = silu(inv1@W_isc1+b)@W_isc2+b ----
    if (t < FDIM) {
        float acc = b_isc1[t];
        for (int k = 0; k < FDIM; ++k) acc += s_inv[k] * W_isc1[k * FDIM + t];
        s_red[t] = acc / (1.f + __expf(-acc));
    }
    __syncthreads();
    if (t < FDIM) {
        float acc = b_isc2[t];
        for (int k = 0; k < FDIM; ++k) acc += s_red[k] * W_isc2[k * FDIM + t];
        s_isc[t] = acc;
    }
    __syncthreads();

    // ---- inv2 = inv1 - isc * sum_d eqf*eqdr ; layernorm ----
    if (t < FDIM) {
        float dot = 0.f;
        #pragma unroll
        for (int d = 0; d < 3; ++d) dot += s_eqfn[d * FDIM + t] * s_eqdrn[d * FDIM + t];
        s_red[t] = s_inv[t] - s_isc[t] * dot;
        s_h[t] = s_red[t];
    }
    __syncthreads();
    for (int s = 64; s > 0; s >>= 1) { if (t < s) s_h[t] += s_h[t + s]; __syncthreads(); }
    const float mu = s_h[0] * (1.f / 128.f);
    __syncthreads();
    if (t < FDIM) { const float d0 = s_red[t] - mu; s_h[t] = d0 * d0; }
    __syncthreads();
    for (int s = 64; s > 0; s >>= 1) { if (t < s) s_h[t] += s_h[t + s]; __syncthreads(); }
    const float var = s_h[0] * (1.f / 128.f);
    if (t < FDIM)
        inv[g * FDIM + t] = (s_red[t] - mu) * rsqrtf(var + 1e-5f) * ln_g[t] + ln_b[t];
}

// ---------------------------------------------------------------------------
extern "C" void kernel_launch(void* const* d_in, const int* in_sizes, int n_in,
                              void* d_out, int out_size, void* d_ws, size_t ws_size,
                              hipStream_t stream)
{
    (void)in_sizes; (void)n_in; (void)out_size; (void)ws_size;

    const int*   z     = (const int*)  d_in[0];
    const float* pos   = (const float*)d_in[1];
    const int*   nbrs  = (const int*)  d_in[2];
    const float* mask  = (const float*)d_in[3];
    const float* emb   = (const float*)d_in[4];
    const float* W_ime = (const float*)d_in[5];
    const float* b_ime = (const float*)d_in[6];
    const float* W_imn1= (const float*)d_in[7];
    const float* b_imn1= (const float*)d_in[8];
    const float* W_imn2= (const float*)d_in[9];
    const float* b_imn2= (const float*)d_in[10];
    const float* W_emc = (const float*)d_in[11];
    const float* W_emf1= (const float*)d_in[12];
    const float* b_emf1= (const float*)d_in[13];
    const float* W_emf2= (const float*)d_in[14];
    const float* b_emf2= (const float*)d_in[15];
    const float* W_esc1= (const float*)d_in[16];
    const float* b_esc1= (const float*)d_in[17];
    const float* W_esc2= (const float*)d_in[18];
    const float* b_esc2= (const float*)d_in[19];
    const float* W_eme1= (const float*)d_in[20];
    const float* W_eme2= (const float*)d_in[21];
    const float* W_isc1= (const float*)d_in[22];
    const float* b_isc1= (const float*)d_in[23];
    const float* W_isc2= (const float*)d_in[24];
    const float* b_isc2= (const float*)d_in[25];
    const float* ln_g  = (const float*)d_in[26];
    const float* ln_b  = (const float*)d_in[27];

    const int NA = BATCH * NATOM;                // 2048 atoms
    float* ws    = (float*)d_ws;
    float* inv   = ws;                           // NA*128
    float* imn   = inv   + (size_t)NA * FDIM;
    float* eqF   = imn   + (size_t)NA * FDIM;
    float* eqf   = eqF   + (size_t)NA * 3;
    float* eqdr0 = eqf   + (size_t)NA * 3 * FDIM;
    float* eqdr1 = eqdr0 + (size_t)NA * 3 * FDIM;
    float* edge  = eqdr1 + (size_t)NA * 3 * FDIM;
    float* cut   = edge  + (size_t)NA * NNBR * NBAS;
    float* dvec  = cut   + (size_t)NA * NNBR;

    k_init<<<NA, 128, 0, stream>>>(z, pos, nbrs, emb, inv, eqF, eqf,
                                   eqdr0, eqdr1, edge, cut, dvec);

    float* dprev = eqdr0;
    float* dnext = eqdr1;
    for (int l = 0; l < NLAY; ++l) {
        const size_t wL = (size_t)l * FDIM * FDIM;   // 128x128 weight stride
        const size_t vL = (size_t)l * FDIM;          // 128-vector stride
        k_imn<<<NA / 32, 256, 0, stream>>>(inv, W_imn1 + wL, b_imn1 + vL,
                                           W_imn2 + wL, b_imn2 + vL, imn);
        k_layer<<<NA, 256, 0, stream>>>(
            nbrs, mask, edge, cut, dvec, imn, inv, eqF, eqf,
            dprev, dnext,
            W_ime + (size_t)l * NBAS * FDIM, b_ime + vL, W_emc + vL,
            W_emf1 + wL, b_emf1 + vL, W_emf2 + wL, b_emf2 + vL,
            W_eme1 + wL, W_eme2 + wL,
            W_esc1 + wL, b_esc1 + vL, W_esc2 + wL, b_esc2 + vL,
            W_isc1 + wL, b_isc1 + vL, W_isc2 + wL, b_isc2 + vL,
            ln_g + vL, ln_b + vL);
        float* tmp = dprev; dprev = dnext; dnext = tmp;
    }

    // outputs: inv (NA*128), eqF (NA*3), eqf (NA*384), eqdr (NA*384)
    float* out = (float*)d_out;
    hipMemcpyAsync(out,                         inv,   (size_t)NA * FDIM     * sizeof(float), hipMemcpyDeviceToDevice, stream);
    hipMemcpyAsync(out + (size_t)NA * FDIM,     eqF,   (size_t)NA * 3        * sizeof(float), hipMemcpyDeviceToDevice, stream);
    hipMemcpyAsync(out + (size_t)NA * (FDIM+3), eqf,   (size_t)NA * 3 * FDIM * sizeof(float), hipMemcpyDeviceToDevice, stream);
    hipMemcpyAsync(out + (size_t)NA * (FDIM+3+3*FDIM), dprev,
                   (size_t)NA * 3 * FDIM * sizeof(float), hipMemcpyDeviceToDevice, stream);
}